// ForceField_33148557590854
// MI455X (gfx1250) — compile-verified
//
#include <hip/hip_runtime.h>

// ---------------------------------------------------------------------------
// ForceField energy on MI455X (gfx1250, wave32).
//   V = sum_bonds kb*(|xi-xj|-b0)^2  +  sum_{i<j} eps_ij*(rod^12 - 2 rod^6)
//
// Roofline: streaming eps_ij/rmin_ij (2 x 256MB fp32, upper triangle ~256MB
// read) dominates -> ~11us at 23.3 TB/s. Loads use non-temporal hints (no
// reuse, working set > 192MB L2). Distance tiles come from one
// V_WMMA_F32_16X16X4_F32 per 16x16 tile:
//   A[m] = ( x_i, y_i, z_i, |x_i|^2 ),  B[:,n] = ( -2x_j, -2y_j, -2z_j, 1 ),
//   C[m][n] = |x_j|^2   =>   D[m][n] = |x_i - x_j|^2
// Per-element LJ math uses v_rcp_f32 (TRANS, ~1ulp, co-executes with VALU)
// instead of the ~10-instruction IEEE division ladder to keep the kernel
// memory-bound rather than VALU-bound.
// ---------------------------------------------------------------------------

typedef __attribute__((ext_vector_type(2))) float v2f;
typedef __attribute__((ext_vector_type(8))) float v8f;

#define N_ATOMS      8192
#define TILE         16
#define NTILE        (N_ATOMS / TILE)   // 512 tiles per dimension
#define WAVES_PER_BLK 8                 // 256 threads = 8 wave32
#define TJG          (NTILE / WAVES_PER_BLK) // 64 tile-groups along j

__global__ void ff_zero_out(float* out) { out[0] = 0.0f; }

// ------------------------- harmonic bond term ------------------------------
__global__ __launch_bounds__(256) void ff_bond_kernel(
    const float* __restrict__ x, const int* __restrict__ pairs,
    const float* __restrict__ kb, const float* __restrict__ b0,
    float* __restrict__ out, int nb)
{
    int b = blockIdx.x * blockDim.x + threadIdx.x;
    float e = 0.0f;
    if (b < nb) {
        int i = pairs[2 * b + 0];
        int j = pairs[2 * b + 1];
        float dx = x[3 * i + 0] - x[3 * j + 0];
        float dy = x[3 * i + 1] - x[3 * j + 1];
        float dz = x[3 * i + 2] - x[3 * j + 2];
        float dis = sqrtf(dx * dx + dy * dy + dz * dz);
        float d = dis - b0[b];
        e = kb[b] * d * d;
    }
    #pragma unroll
    for (int off = 16; off > 0; off >>= 1) e += __shfl_down(e, off, 32);
    __shared__ float wsum[WAVES_PER_BLK];
    int lane = threadIdx.x & 31, wave = threadIdx.x >> 5;
    if (lane == 0) wsum[wave] = e;
    __syncthreads();
    if (threadIdx.x == 0) {
        float s = 0.0f;
        #pragma unroll
        for (int w = 0; w < WAVES_PER_BLK; ++w) s += wsum[w];
        atomicAdd(out, s);
    }
}

// ------------------------- all-pairs LJ term -------------------------------
// One wave computes one 16x16 tile (ti,tj) of the upper triangle.
// blockIdx.x = ti * TJG + tjg ; wave w handles tj = tjg*8 + w, so the 8 waves
// of a block read adjacent 64B column segments of the same eps/rmin rows
// (full cacheline utilization).
__global__ __launch_bounds__(256) void ff_lj_kernel(
    const float* __restrict__ x,
    const float* __restrict__ eps,
    const float* __restrict__ rmin,
    float* __restrict__ out)
{
    const int lane = threadIdx.x & 31;
    const int wave = threadIdx.x >> 5;
    const int ti   = blockIdx.x / TJG;
    const int tjg  = blockIdx.x % TJG;
    const int tj   = tjg * WAVES_PER_BLK + wave;

    __shared__ float wsum[WAVES_PER_BLK];
    float lsum = 0.0f;

    if (tj >= ti) {                       // wave-uniform: EXEC stays all-1s
        const int n15 = lane & 15;        // tile row/col index held by lane
        const int hi  = lane >> 4;        // 0: lanes 0-15, 1: lanes 16-31

        // --- load the 16 i-atoms and 16 j-atoms of this tile ---
        const int ai = ti * TILE + n15;
        const int bj = tj * TILE + n15;
        float xi = x[3 * ai + 0], yi = x[3 * ai + 1], zi = x[3 * ai + 2];
        float sqi = xi * xi + yi * yi + zi * zi;
        float xj = x[3 * bj + 0], yj = x[3 * bj + 1], zj = x[3 * bj + 2];
        float sqj = xj * xj + yj * yj + zj * zj;

        // --- A fragment: 16x4 f32, K = (x, y, z, |x|^2) ---
        // ISA layout: lanes 0-15 hold K0 (v0) / K1 (v1); lanes 16-31 hold K2/K3.
        v2f afrag;
        afrag.x = hi ? zi  : xi;
        afrag.y = hi ? sqi : yi;

        // --- B fragment: 4x16 f32, column n = (-2x_j, -2y_j, -2z_j, 1) ---
        v2f bfrag;
        bfrag.x = hi ? (-2.0f * zj) : (-2.0f * xj);
        bfrag.y = hi ? 1.0f         : (-2.0f * yj);

        // --- C fragment: C[m][n] = |x_j|^2 (broadcast over rows) ---
        v8f cfrag;
        #pragma unroll
        for (int r = 0; r < 8; ++r) cfrag[r] = sqj;

        // D[m][n] = |x_i - x_j|^2 for the whole 16x16 tile in one WMMA
        v8f d2 = __builtin_amdgcn_wmma_f32_16x16x4_f32(
            /*neg_a=*/false, afrag, /*neg_b=*/false, bfrag,
            /*c_mod=*/(short)0, cfrag, /*reuse_a=*/false, /*reuse_b=*/false);

        // C/D layout: VGPR r <-> row m = r + 8*hi, column n = lane&15.
        // eps/rmin stream once from HBM -> non-temporal loads.
        const size_t rowbase = (size_t)(ti * TILE + hi * 8) * N_ATOMS
                             + (size_t)(tj * TILE + n15);
        const float* ep = eps  + rowbase;
        const float* rp = rmin + rowbase;

        #pragma unroll
        for (int r = 0; r < 8; ++r) {
            const int m  = r + hi * 8;
            float e  = __builtin_nontemporal_load(ep + (size_t)r * N_ATOMS);
            float rm = __builtin_nontemporal_load(rp + (size_t)r * N_ATOMS);
            float dd   = fmaxf(d2[r], 1e-12f);        // guard diag/FP noise
            float inv  = __builtin_amdgcn_rcpf(dd);   // v_rcp_f32, ~1ulp TRANS
            float rod2 = (rm * rm) * inv;             // (rmin/dist)^2, no sqrt
            float r6   = rod2 * rod2 * rod2;          // rod^6
            float v    = e * r6 * (r6 - 2.0f);        // eps*(rod^12 - 2 rod^6)
            bool keep  = (ti < tj) || (m < n15);      // strict upper triangle
            lsum += keep ? v : 0.0f;
        }
    }

    // --- reduction: wave shuffle -> LDS -> one atomic per block ---
    #pragma unroll
    for (int off = 16; off > 0; off >>= 1) lsum += __shfl_down(lsum, off, 32);
    if (lane == 0) wsum[wave] = lsum;
    __syncthreads();
    if (threadIdx.x == 0) {
        float s = 0.0f;
        #pragma unroll
        for (int w = 0; w < WAVES_PER_BLK; ++w) s += wsum[w];
        atomicAdd(out, s);
    }
}

// ---------------------------------------------------------------------------
extern "C" void kernel_launch(void* const* d_in, const int* in_sizes, int n_in,
                              void* d_out, int out_size, void* d_ws, size_t ws_size,
                              hipStream_t stream) {
    const float* x     = (const float*)d_in[0];   // [8192,3] f32
    const int*   pairs = (const int*)  d_in[1];   // [8192,2] i32
    const float* kb    = (const float*)d_in[2];   // [8192]   f32
    const float* b0    = (const float*)d_in[3];   // [8192]   f32
    const float* eps   = (const float*)d_in[4];   // [8192,8192] f32
    const float* rmin  = (const float*)d_in[5];   // [8192,8192] f32
    float* out = (float*)d_out;                   // scalar f32

    const int nb = in_sizes[2];                   // number of bonds (8192)

    ff_zero_out<<<1, 1, 0, stream>>>(out);
    ff_bond_kernel<<<(nb + 255) / 256, 256, 0, stream>>>(x, pairs, kb, b0, out, nb);
    ff_lj_kernel<<<NTILE * TJG, 256, 0, stream>>>(x, eps, rmin, out);
}